// GINConv_19731079758624
// MI455X (gfx1250) — compile-verified
//
#include <hip/hip_runtime.h>
#include <hip/hip_bf16.h>

// ---------------------------------------------------------------------------
// GINConv (EPS=0):  out = relu((x + scatter_add(x[src] -> dst)) @ W1 + b1) @ W2 + b2
// N=100000 nodes, D=128, E=1600000 edges. All fp32.
// ---------------------------------------------------------------------------

#define DIM        128
#define TILES_K    32          // 128 / 4 (K per V_WMMA_F32_16X16X4_F32)
#define LDS_STRIDE 132         // 128 + 4 pad: banks (4*m + k) % 64 -> conflict-free

typedef float v2f __attribute__((ext_vector_type(2)));
typedef float v8f __attribute__((ext_vector_type(8)));

// -------------------------------------------------------------------- init --
// h0 = x  (self term; EPS == 0)
__global__ void gin_init_copy(const float* __restrict__ x,
                              float* __restrict__ h0, int total4) {
    int i = blockIdx.x * blockDim.x + threadIdx.x;
    if (i < total4) {
        ((float4*)h0)[i] = ((const float4*)x)[i];
    }
}

// ----------------------------------------------------------------- scatter --
// One wave (32 lanes) per edge; each lane owns 4 consecutive feature dims.
// agg[dst] += x[src] via hardware fp32 atomics (L2-resident, fire-and-forget).
__global__ void gin_scatter_add(const float* __restrict__ x,
                                const int* __restrict__ ei,   // [2, E] flat
                                float* __restrict__ h0, int nEdges) {
    int gtid = blockIdx.x * blockDim.x + threadIdx.x;
    int wave = gtid >> 5;
    int lane = gtid & 31;
    if (wave >= nEdges) return;

    int s = ei[wave];            // src row
    int d = ei[nEdges + wave];   // dst row

    float4 v = ((const float4*)(x + (size_t)s * DIM))[lane];
    float* dp = h0 + (size_t)d * DIM + lane * 4;
    unsafeAtomicAdd(dp + 0, v.x);
    unsafeAtomicAdd(dp + 1, v.y);
    unsafeAtomicAdd(dp + 2, v.z);
    unsafeAtomicAdd(dp + 3, v.w);
}

// -------------------------------------------------------- async tile stage --
// Stage one 16x128 fp32 tile (2048 floats, padded rows) global -> LDS via the
// CDNA5 async-to-LDS path (ASYNCcnt, no VGPR round-trip). 256 threads issue
// 2 x b128 transfers each. Low 32 bits of a generic LDS pointer == LDS offset.
__device__ __forceinline__ void async_tile_load(const float* __restrict__ src,
                                                float* ldsbuf, int m0, int tid) {
#pragma unroll
    for (int i = 0; i < 2; ++i) {
        const int f  = tid + i * 256;  // 0..511 float4 slots
        const int r  = f >> 5;         // row 0..15
        const int c4 = f & 31;         // float4 column
        unsigned int ldst =
            (unsigned int)(unsigned long long)(uintptr_t)&ldsbuf[r * LDS_STRIDE + c4 * 4];
        unsigned long long g =
            (unsigned long long)(uintptr_t)(src + (size_t)(m0 + r) * DIM + c4 * 4);
        asm volatile("global_load_async_to_lds_b128 %0, %1, off"
                     :: "v"(ldst), "v"(g) : "memory");
    }
}

// --------------------------------------------------------------------- mlp --
// Block = 256 threads = 8 waves. Each wave owns a 16-wide output column tile
// (n0 = 16*wave) for BOTH layers; W1/W2 B-fragments live in registers for the
// whole block lifetime. A-tiles are double-buffered: the async DMA for tile
// i+1 overlaps the 64 WMMAs of tile i.
//
// Fragment layouts per CDNA5 ISA 7.12.2 (wave32):
//   A 16x4 f32 : lanes 0-15 -> M=lane, {K, K+1}; lanes 16-31 -> M=lane-16, {K+2, K+3}
//   B 4x16 f32 : lanes 0-15 -> N=lane, rows {K, K+1}; lanes 16-31 -> rows {K+2, K+3}
//   C/D 16x16  : VGPR r -> M = r + (lane>=16 ? 8 : 0), N = lane & 15
__global__ __launch_bounds__(256)
void gin_mlp(const float* __restrict__ h0,
             const float* __restrict__ W1, const float* __restrict__ b1,
             const float* __restrict__ W2, const float* __restrict__ b2,
             float* __restrict__ out, int nTiles) {
    __shared__ float At[2][16 * LDS_STRIDE];   // double-buffered input tiles
    __shared__ float H1[16 * LDS_STRIDE];      // layer-1 activations

    const int tid  = threadIdx.x;
    const int lane = tid & 31;
    const int w    = tid >> 5;          // wave id 0..7
    const int n0   = w * 16;            // this wave's output column base
    const int col  = lane & 15;
    const int hi   = lane >> 4;         // 0: K,K+1 half | 1: K+2,K+3 half

    // Kick off the first tile's DMA before the (long) weight preload so the
    // two overlap.
    const int mt0 = (int)blockIdx.x;
    if (mt0 < nTiles) async_tile_load(h0, At[0], mt0 * 16, tid);

    // ---- preload B fragments for both layers into registers ----
    v2f B1r[TILES_K], B2r[TILES_K];
#pragma unroll
    for (int s = 0; s < TILES_K; ++s) {
        const int kr = s * 4 + hi * 2;
        B1r[s].x = W1[(size_t)(kr    ) * DIM + n0 + col];
        B1r[s].y = W1[(size_t)(kr + 1) * DIM + n0 + col];
        B2r[s].x = W2[(size_t)(kr    ) * DIM + n0 + col];
        B2r[s].y = W2[(size_t)(kr + 1) * DIM + n0 + col];
    }
    const float bias1 = b1[n0 + col];
    const float bias2 = b2[n0 + col];

    int cur = 0;
    for (int mt = mt0; mt < nTiles; mt += gridDim.x) {
        const int m0 = mt * 16;   // N = 100000 is an exact multiple of 16

        // My async transfers landed; barrier => everyone's landed AND everyone
        // is done reading At[cur^1] + H1 from the previous iteration.
        asm volatile("s_wait_asynccnt 0x0" ::: "memory");
        __syncthreads();

        const int nxt = mt + (int)gridDim.x;
        if (nxt < nTiles) async_tile_load(h0, At[cur ^ 1], nxt * 16, tid);

        // ---- layer 1: acc = A @ W1[:, n0:n0+16] ----
        const float* Ab = &At[cur][col * LDS_STRIDE + hi * 2];
        v8f acc = {0.f, 0.f, 0.f, 0.f, 0.f, 0.f, 0.f, 0.f};
#pragma unroll
        for (int c = 0; c < 4; ++c) {          // chunks: 4 ds_loads -> 8 WMMAs
            v2f a[8];
#pragma unroll
            for (int s = 0; s < 8; ++s) {
                const float* ap = Ab + (c * 8 + s) * 4;
                a[s].x = ap[0]; a[s].y = ap[1];
            }
#pragma unroll
            for (int s = 0; s < 8; ++s) {
                acc = __builtin_amdgcn_wmma_f32_16x16x4_f32(
                    false, a[s], false, B1r[c * 8 + s], (short)0, acc, false, false);
            }
        }

        // ---- bias + ReLU -> H1 tile in LDS ----
#pragma unroll
        for (int r = 0; r < 8; ++r) {
            float v = acc[r] + bias1;
            v = v > 0.f ? v : 0.f;
            H1[(r + hi * 8) * LDS_STRIDE + n0 + col] = v;
        }
        __syncthreads();

        // ---- layer 2: acc2 = H1 @ W2[:, n0:n0+16] ----
        const float* Hb = &H1[col * LDS_STRIDE + hi * 2];
        v8f acc2 = {0.f, 0.f, 0.f, 0.f, 0.f, 0.f, 0.f, 0.f};
#pragma unroll
        for (int c = 0; c < 4; ++c) {
            v2f a[8];
#pragma unroll
            for (int s = 0; s < 8; ++s) {
                const float* ap = Hb + (c * 8 + s) * 4;
                a[s].x = ap[0]; a[s].y = ap[1];
            }
#pragma unroll
            for (int s = 0; s < 8; ++s) {
                acc2 = __builtin_amdgcn_wmma_f32_16x16x4_f32(
                    false, a[s], false, B2r[c * 8 + s], (short)0, acc2, false, false);
            }
        }

        // ---- bias + store ----
#pragma unroll
        for (int r = 0; r < 8; ++r) {
            out[(size_t)(m0 + r + hi * 8) * DIM + n0 + col] = acc2[r] + bias2;
        }

        cur ^= 1;
    }
}

// ---------------------------------------------------------------------------
extern "C" void kernel_launch(void* const* d_in, const int* in_sizes, int n_in,
                              void* d_out, int out_size, void* d_ws, size_t ws_size,
                              hipStream_t stream) {
    const float* x  = (const float*)d_in[0];
    const int*   ei = (const int*)  d_in[1];   // int32 edge indices, [2, E] flat
    const float* W1 = (const float*)d_in[2];
    const float* b1 = (const float*)d_in[3];
    const float* W2 = (const float*)d_in[4];
    const float* b2 = (const float*)d_in[5];
    float* out = (float*)d_out;

    const int nNodes = in_sizes[0] / DIM;  // 100000
    const int nEdges = in_sizes[1] / 2;    // 1600000

    // Aggregation buffer: prefer workspace; fall back to in-place on d_out
    // (safe: each MLP block reads & writes a disjoint 16-row tile).
    const size_t need = (size_t)nNodes * DIM * sizeof(float);
    float* h0 = (ws_size >= need) ? (float*)d_ws : out;

    // 1) h0 = x
    {
        const int total4 = nNodes * (DIM / 4);
        const int blocks = (total4 + 255) / 256;
        gin_init_copy<<<blocks, 256, 0, stream>>>(x, h0, total4);
    }
    // 2) h0[dst] += x[src] for every edge
    {
        const long long threads = (long long)nEdges * 32;
        const int blocks = (int)((threads + 255) / 256);
        gin_scatter_add<<<blocks, 256, 0, stream>>>(x, ei, h0, nEdges);
    }
    // 3) out = relu(h0 @ W1 + b1) @ W2 + b2
    {
        const int nTiles = (nNodes + 15) / 16;           // 6250
        const int blocks = nTiles < 1024 ? nTiles : 1024; // W regs amortized ~6x
        gin_mlp<<<blocks, 256, 0, stream>>>(h0, W1, b1, W2, b2, out, nTiles);
    }
}